// ThreeStateExplorer_64561948393588
// MI455X (gfx1250) — compile-verified
//
#include <hip/hip_runtime.h>
#include <hip/hip_bf16.h>
#include <math.h>
#include <stdint.h>

#ifndef USE_ASYNC_LDS
#define USE_ASYNC_LDS 1
#endif

#if defined(__has_builtin)
#if __has_builtin(__builtin_amdgcn_wmma_f32_16x16x4_f32)
#define HAVE_WMMA_SCAN 1
#endif
#endif
#ifndef HAVE_WMMA_SCAN
#define HAVE_WMMA_SCAN 0
#endif

#define BPART 512
#define NSTEP 20000
#define NTUM  64
#define TPB   256
#define CHUNK 80        // 250 active threads * 80 = 20000 exactly

#define RATE    1.0e-3f
#define SPEED0  1.0e-3f
#define SPEED1  5.0e-3f
#define PA_SIG  0.1f
#define NP_SIG  0.5f
#define PI_F    3.14159265358979323846f
#define MAP_ID  0x24u   // identity map: 0->0, 1->1, 2->2

typedef float v2f __attribute__((ext_vector_type(2)));
typedef float v8f __attribute__((ext_vector_type(8)));

struct F3 { float x, y, z; };

__device__ __forceinline__ float dot3(F3 a, F3 b){ return a.x*b.x + a.y*b.y + a.z*b.z; }
__device__ __forceinline__ F3 cross3(F3 a, F3 b){
  F3 r; r.x = a.y*b.z - a.z*b.y; r.y = a.z*b.x - a.x*b.z; r.z = a.x*b.y - a.y*b.x; return r;
}
__device__ __forceinline__ F3 norm3(F3 a){
  float n = sqrtf(dot3(a,a));
  F3 r; r.x = a.x/n; r.y = a.y/n; r.z = a.z/n; return r;
}
__device__ __forceinline__ F3 ortho3(F3 v, F3 u){
  float f = dot3(v,u) / dot3(u,u);
  F3 r; r.x = v.x - f*u.x; r.y = v.y - f*u.y; r.z = v.z - f*u.z; return r;
}
// Rodrigues: R(u,ang) w = c*w + s*(u x w) + (1-c)*(u.w)*u
__device__ __forceinline__ F3 rot_axis(F3 u, float c, float s, F3 w){
  F3 x = cross3(u, w);
  float ud = dot3(u, w);
  float oc = 1.0f - c;
  F3 r;
  r.x = c*w.x + s*x.x + oc*ud*u.x;
  r.y = c*w.y + s*x.y + oc*ud*u.y;
  r.z = c*w.z + s*x.z + oc*ud*u.z;
  return r;
}
__device__ __forceinline__ float cauchy_angle(float u, float sig){
  float a = sig * tanf(PI_F * (u - 0.5f));
  unsigned bits = __float_as_uint(a);
  if ((bits & 0x7f800000u) == 0x7f800000u) a = 0.0f;   // nan/inf -> 0
  return fmodf(a, PI_F);
}
// compose transition maps: result(s) = later(earlier(s)); 2-bit entries packed in 6 bits
__device__ __forceinline__ unsigned mcompose(unsigned later, unsigned earlier){
  unsigned r;
  r  =  (later >> (2u * ( earlier        & 3u))) & 3u;
  r |= ((later >> (2u * ((earlier >> 2u) & 3u))) & 3u) << 2u;
  r |= ((later >> (2u * ((earlier >> 4u) & 3u))) & 3u) << 4u;
  return r;
}

#if HAVE_WMMA_SCAN
// Inclusive within-row prefix of a 16x16 matrix S (row-major in LDS, sc[16m+j]),
// computed as P = S * U (U upper-triangular ones) with 4 chained
// v_wmma_f32_16x16x4_f32 (K=4 slices). One wave32, EXEC all ones.
// A 16x4 f32 layout (ISA 7.12.2): lane(0-15)=M, lanes 16-31 repeat M with K=2,3;
// VGPR0 holds K={0|2}, VGPR1 holds K={1|3}. C/D: lane-half selects M / M+8.
__device__ __forceinline__ void wmma_row_scan(const float* __restrict__ sc,
                                              float* __restrict__ sp, int lane) {
  const int half = lane >> 4;        // 0: K=0,1  1: K=2,3 (A); M vs M+8 (D)
  const int lm   = lane & 15;        // M for A, N for B/D
  v8f acc = {};
#pragma unroll
  for (int q = 0; q < 4; ++q) {
    v2f a, b;
    a[0] = sc[16*lm + 4*q + 2*half + 0];
    a[1] = sc[16*lm + 4*q + 2*half + 1];
    b[0] = ((4*q + 2*half + 0) <= lm) ? 1.0f : 0.0f;   // U slice rows 4q..4q+3
    b[1] = ((4*q + 2*half + 1) <= lm) ? 1.0f : 0.0f;
    acc = __builtin_amdgcn_wmma_f32_16x16x4_f32(false, a, false, b,
                                                (short)0, acc, false, false);
  }
#pragma unroll
  for (int r = 0; r < 8; ++r)
    sp[(r + 8*half)*16 + lm] = acc[r];
}
#endif

// ---------------- Kernel 1: per-particle tumble-frame chain (64 sequential Rodrigues steps)
__global__ void frames_kernel(const float* __restrict__ v0g, const float* __restrict__ v1g,
                              const float* __restrict__ upg, const float* __restrict__ ung,
                              float* __restrict__ fr) {
  int b = blockIdx.x * blockDim.x + threadIdx.x;
  if (b >= BPART) return;
  F3 v0; v0.x = v0g[3*b+0]; v0.y = v0g[3*b+1]; v0.z = v0g[3*b+2];
  F3 v1; v1.x = v1g[3*b+0]; v1.y = v1g[3*b+1]; v1.z = v1g[3*b+2];

  F3 e0 = norm3(v0);
  F3 e1 = norm3(ortho3(v1, e0));
  F3 e2 = norm3(cross3(e0, e1));

  float* out = fr + (size_t)b * (NTUM + 1) * 3;
  out[0] = e0.x; out[1] = e0.y; out[2] = e0.z;

  for (int k = 0; k < NTUM; ++k) {
    float ap = cauchy_angle(upg[(size_t)b*NTUM + k], PA_SIG);
    float an = cauchy_angle(ung[(size_t)b*NTUM + k], NP_SIG);
    // planar: rotate about e2
    float c = cosf(ap), s = sinf(ap);
    F3 ne0 = rot_axis(e2, c, s, e0);
    F3 ne1 = rot_axis(e2, c, s, e1);
    e0 = norm3(ne0);
    e1 = norm3(ne1);
    e1 = norm3(ortho3(e1, e0));
    e2 = norm3(cross3(e0, e1));
    // nonplanar: rotate about e1 (only e0 is rotated, per reference)
    c = cosf(an); s = sinf(an);
    e0 = norm3(rot_axis(e1, c, s, e0));
    e1 = norm3(ortho3(e1, e0));
    e2 = norm3(cross3(e0, e1));
    out[(k+1)*3+0] = e0.x; out[(k+1)*3+1] = e0.y; out[(k+1)*3+2] = e0.z;
  }
}

// ---------------- Kernel 2: one workgroup per particle; 3 chained block scans + output
__global__ __launch_bounds__(TPB)
void traj_kernel(const float* __restrict__ x0g, const float* __restrict__ r0g,
                 const float* __restrict__ r1g, const float* __restrict__ fr,
                 float* __restrict__ X) {
  const int b   = blockIdx.x;
  const int tid = threadIdx.x;

  __shared__ unsigned char smaps[TPB * CHUNK];     // 20 KB of 6-bit transition maps
  __shared__ unsigned sscan[TPB];
  __shared__ float sx[TPB], sy[TPB], sz[TPB];      // per-thread chunk sums (S matrix)
  __shared__ float px[TPB], py[TPB], pz[TPB];      // row-prefixed S (WMMA output)
  __shared__ float sframes[(NTUM + 1) * 3];        // 195 floats

  // ---- stage frame table to LDS (CDNA5 async global->LDS path) ----
  {
    const float* fb = fr + (size_t)b * (NTUM + 1) * 3;
    if (tid < (NTUM + 1) * 3) {
#if USE_ASYNC_LDS
      unsigned lds = (unsigned)(uintptr_t)(&sframes[tid]);   // LDS byte offset (flat addr low 32b)
      unsigned off = (unsigned)tid * 4u;
      asm volatile("global_load_async_to_lds_b32 %0, %1, %2"
                   :: "v"(lds), "v"(off), "s"(fb) : "memory");
#else
      sframes[tid] = fb[tid];
#endif
    }
#if USE_ASYNC_LDS
    asm volatile("s_wait_asynccnt 0x0" ::: "memory");
#endif
  }

  const float x0x = x0g[3*b+0], x0y = x0g[3*b+1], x0z = x0g[3*b+2];

  const int  base   = tid * CHUNK;
  const bool active = base < NSTEP;

  // ---- Pass A: stream r0/r1 once, emit per-step transition map to LDS, fold chunk map ----
  unsigned cm = MAP_ID;
  if (active) {
    const float4* p0 = (const float4*)(r0g + (size_t)b * NSTEP + base);
    const float4* p1 = (const float4*)(r1g + (size_t)b * NSTEP + base);
    unsigned char* mp = smaps + base;
    const float hi = 1.0f - RATE;
    for (int q = 0; q < CHUNK/4; ++q) {
      if ((q & 7) == 0) {                       // one cacheline ahead
        __builtin_prefetch(p0 + q + 8, 0, 3);   // -> global_prefetch_b8
        __builtin_prefetch(p1 + q + 8, 0, 3);
      }
      float4 a = p0[q];
      float4 c4 = p1[q];
      float r0v[4] = {a.x, a.y, a.z, a.w};
      float r1v[4] = {c4.x, c4.y, c4.z, c4.w};
#pragma unroll
      for (int e = 0; e < 4; ++e) {
        unsigned f0  = (r0v[e] < RATE) ? 1u : ((r0v[e] > hi) ? 2u : 0u);
        unsigned f12 = (r1v[e] < RATE) ? 0u : 1u;
        unsigned m = f0 | (f12 << 2) | (f12 << 4);
        mp[4*q + e] = (unsigned char)m;
        cm = mcompose(m, cm);
      }
    }
  }
  __syncthreads();

  // ---- Scan 1: map composition (non-commutative, ordered Hillis-Steele) ----
  sscan[tid] = cm;
  __syncthreads();
  for (int d = 1; d < TPB; d <<= 1) {
    unsigned v = sscan[tid];
    unsigned w = (tid >= d) ? sscan[tid - d] : MAP_ID;
    __syncthreads();
    sscan[tid] = mcompose(v, w);
    __syncthreads();
  }
  unsigned excl = (tid == 0) ? MAP_ID : sscan[tid - 1];
  int s_in = (int)(excl & 3u);            // incoming state = excl applied to s0=0
  __syncthreads();

  // ---- Pass B: local tumble count ----
  int cnt = 0;
  if (active) {
    int s = s_in;
    const unsigned char* mp = smaps + base;
    for (int j = 0; j < CHUNK; ++j) {
      s = (mp[j] >> (2*s)) & 3;
      cnt += (s == 2 && (base + j) != 0) ? 1 : 0;
    }
  }
  // ---- Scan 2: integer add scan of tumble counts ----
  sscan[tid] = (unsigned)cnt;
  __syncthreads();
  for (int d = 1; d < TPB; d <<= 1) {
    unsigned v = sscan[tid];
    unsigned w = (tid >= d) ? sscan[tid - d] : 0u;
    __syncthreads();
    sscan[tid] = v + w;
    __syncthreads();
  }
  int cbase = (tid == 0) ? 0 : (int)sscan[tid - 1];
  __syncthreads();

  // ---- Pass C: local float3 velocity sum ----
  float sxv = 0.f, syv = 0.f, szv = 0.f;
  if (active) {
    int s = s_in, c = cbase;
    const unsigned char* mp = smaps + base;
    for (int j = 0; j < CHUNK; ++j) {
      s = (mp[j] >> (2*s)) & 3;
      int step = base + j;
      bool tum = (s == 2) && (step != 0);
      c += tum ? 1 : 0;
      float vx, vy, vz;
      if (step == 0)      { vx = x0x; vy = x0y; vz = x0z; }
      else if (tum)       { vx = 0.f; vy = 0.f; vz = 0.f; }
      else {
        int idx = c - 1; idx = idx < 0 ? 0 : (idx > NTUM ? NTUM : idx);
        float sp = (s == 1) ? SPEED1 : SPEED0;
        vx = sframes[3*idx+0]*sp; vy = sframes[3*idx+1]*sp; vz = sframes[3*idx+2]*sp;
      }
      sxv += vx; syv += vy; szv += vz;
    }
  }
  // ---- Scan 3: float3 scan over 256 chunk sums ----
  sx[tid] = sxv; sy[tid] = syv; sz[tid] = szv;
  __syncthreads();
#if HAVE_WMMA_SCAN
  // Prefix-as-triangular-matmul: P = S*U via v_wmma_f32_16x16x4_f32 (wave 0 only).
  if (tid < 32) {
    wmma_row_scan(sx, px, tid);
    wmma_row_scan(sy, py, tid);
    wmma_row_scan(sz, pz, tid);
  }
  __syncthreads();
  float ax = 0.f, ay = 0.f, az = 0.f;
  if (tid > 0) {
    const int t2 = tid - 1, i2 = t2 >> 4, j2 = t2 & 15;
    ax = px[i2*16 + j2]; ay = py[i2*16 + j2]; az = pz[i2*16 + j2];
    for (int r = 0; r < i2; ++r) {           // carry: totals of earlier rows
      ax += px[r*16 + 15]; ay += py[r*16 + 15]; az += pz[r*16 + 15];
    }
  }
#else
  for (int d = 1; d < TPB; d <<= 1) {
    float vx = sx[tid], vy = sy[tid], vz = sz[tid];
    float wx = 0.f, wy = 0.f, wz = 0.f;
    if (tid >= d) { wx = sx[tid-d]; wy = sy[tid-d]; wz = sz[tid-d]; }
    __syncthreads();
    sx[tid] = vx + wx; sy[tid] = vy + wy; sz[tid] = vz + wz;
    __syncthreads();
  }
  float ax = 0.f, ay = 0.f, az = 0.f;
  if (tid > 0) { ax = sx[tid-1]; ay = sy[tid-1]; az = sz[tid-1]; }
#endif

  // ---- Pass D: recompute velocities, running prefix, stream X out ----
  if (active) {
    int s = s_in, c = cbase;
    const unsigned char* mp = smaps + base;
    float* out = X + ((size_t)b * NSTEP + base) * 3;
    for (int j = 0; j < CHUNK; ++j) {
      s = (mp[j] >> (2*s)) & 3;
      int step = base + j;
      bool tum = (s == 2) && (step != 0);
      c += tum ? 1 : 0;
      float vx, vy, vz;
      if (step == 0)      { vx = x0x; vy = x0y; vz = x0z; }
      else if (tum)       { vx = 0.f; vy = 0.f; vz = 0.f; }
      else {
        int idx = c - 1; idx = idx < 0 ? 0 : (idx > NTUM ? NTUM : idx);
        float sp = (s == 1) ? SPEED1 : SPEED0;
        vx = sframes[3*idx+0]*sp; vy = sframes[3*idx+1]*sp; vz = sframes[3*idx+2]*sp;
      }
      ax += vx; ay += vy; az += vz;
      out[3*j+0] = ax; out[3*j+1] = ay; out[3*j+2] = az;
    }
  }
}

extern "C" void kernel_launch(void* const* d_in, const int* in_sizes, int n_in,
                              void* d_out, int out_size, void* d_ws, size_t ws_size,
                              hipStream_t stream) {
  const float* x0 = (const float*)d_in[0];
  const float* v0 = (const float*)d_in[1];
  const float* v1 = (const float*)d_in[2];
  const float* r0 = (const float*)d_in[3];
  const float* r1 = (const float*)d_in[4];
  const float* up = (const float*)d_in[5];
  const float* un = (const float*)d_in[6];
  // d_in[7] = T (unused by reference body)
  float* X  = (float*)d_out;
  float* fr = (float*)d_ws;   // (BPART, NTUM+1, 3) frame table

  frames_kernel<<<(BPART + TPB - 1)/TPB, TPB, 0, stream>>>(v0, v1, up, un, fr);
  traj_kernel<<<BPART, TPB, 0, stream>>>(x0, r0, r1, fr, X);

  (void)in_sizes; (void)n_in; (void)out_size; (void)ws_size;
}